// HollywoodFFTTriton_53584011985639
// MI455X (gfx1250) — compile-verified
//
#include <hip/hip_runtime.h>

typedef __attribute__((ext_vector_type(2))) float v2f;
typedef __attribute__((ext_vector_type(8))) float v8f;

#define NFFT 4096

// One radix-16 pass: 2 tiles per wave (t = wave, wave+8), each tile is a
// complex [16x16 DFT] x [16x16 data] product done as 16 chained
// v_wmma_f32_16x16x4_f32 ops, followed by per-element twiddle + LDS store.
template <int P>
__device__ __forceinline__ void fft_pass(
    const float* __restrict__ Rr, const float* __restrict__ Ri,   // read buffers
    float* __restrict__ Qr, float* __restrict__ Qi,               // write buffers
    const float* __restrict__ sWr, const float* __restrict__ sWi, // twiddle table (LDS)
    const v2f* fa_r, const v2f* fa_i, const v2f* fa_n,            // F16 chunks (re, im, -im)
    int wave, int hi, int l15)
{
#pragma unroll
  for (int tt = 0; tt < 2; ++tt) {
    const int t = wave + 8 * tt;

    // Load B operand rows for this lane: rows r = 4c + 2*hi + v, col = l15
    float br[8], bi[8];
#pragma unroll
    for (int c = 0; c < 4; ++c) {
#pragma unroll
      for (int v = 0; v < 2; ++v) {
        const int r = 4 * c + 2 * hi + v;
        int a;
        if (P == 1)      a = r * 256 + t * 16 + l15;  // [n2][n1=t][n0=j]
        else if (P == 2) a = t * 256 + r * 16 + l15;  // [k0=t][n1][n0=j]
        else             a = t * 256 + l15 * 16 + r;  // [k0=t][k1=j][n0]
        br[2 * c + v] = Rr[a];
        bi[2 * c + v] = Ri[a];
      }
    }

    v8f accR = {0.f, 0.f, 0.f, 0.f, 0.f, 0.f, 0.f, 0.f};
    v8f accI = {0.f, 0.f, 0.f, 0.f, 0.f, 0.f, 0.f, 0.f};
#pragma unroll
    for (int c = 0; c < 4; ++c) {
      v2f vbr; vbr.x = br[2 * c]; vbr.y = br[2 * c + 1];
      v2f vbi; vbi.x = bi[2 * c]; vbi.y = bi[2 * c + 1];
      // Re += Fr*Br - Fi*Bi ; Im += Fr*Bi + Fi*Br   (f32 WMMA: no A/B NEG, use -Fi)
      accR = __builtin_amdgcn_wmma_f32_16x16x4_f32(false, fa_r[c], false, vbr, (short)0, accR, false, false);
      accR = __builtin_amdgcn_wmma_f32_16x16x4_f32(false, fa_n[c], false, vbi, (short)0, accR, false, false);
      accI = __builtin_amdgcn_wmma_f32_16x16x4_f32(false, fa_r[c], false, vbi, (short)0, accI, false, false);
      accI = __builtin_amdgcn_wmma_f32_16x16x4_f32(false, fa_i[c], false, vbr, (short)0, accI, false, false);
    }

    // Twiddle + scatter to LDS.  D layout: VGPR r -> M=r (lanes 0-15) / M=r+8, N=l15.
#pragma unroll
    for (int r = 0; r < 8; ++r) {
      const int M = r + 8 * hi;
      float dr = accR[r], di = accI[r];
      if (P == 1) {
        const int idx = (16 * t * M) & 4095;          // w4096^(16*n1*k0), n1=t, k0=M
        const float wr = sWr[idx], wi = sWi[idx];
        const float tr = dr * wr - di * wi;
        di = dr * wi + di * wr;
        dr = tr;
      } else if (P == 2) {
        const int idx = (l15 * (16 * M + t)) & 4095;  // w4096^(n0*(16*k1+k0))
        const float wr = sWr[idx], wi = sWi[idx];
        const float tr = dr * wr - di * wi;
        di = dr * wi + di * wr;
        dr = tr;
      }
      int a;
      if (P == 1)      a = M * 256 + t * 16 + l15;    // [k0][n1=t][n0=j]
      else if (P == 2) a = t * 256 + M * 16 + l15;    // [k0=t][k1][n0=j]
      else             a = M * 256 + l15 * 16 + t;    // k = k2*256 + k1*16 + k0 (natural order)
      Qr[a] = dr;
      Qi[a] = di;
    }
  }
}

extern "C" __global__ void __launch_bounds__(256) fft4096_wmma(
    const float* __restrict__ x_re, const float* __restrict__ x_im,
    const float* __restrict__ W_re, const float* __restrict__ W_im,
    float* __restrict__ out, unsigned long long im_off)
{
  extern __shared__ float smem[];
  float* sAr = smem;
  float* sAi = smem + NFFT;
  float* sBr = smem + 2 * NFFT;
  float* sBi = smem + 3 * NFFT;
  float* sWr = smem + 4 * NFFT;
  float* sWi = smem + 5 * NFFT;

  const int tid  = threadIdx.x;
  const int lane = tid & 31;
  const int wave = tid >> 5;
  const int hi   = lane >> 4;
  const int l15  = lane & 15;
  const size_t row = blockIdx.x;

  // Coalesced stage: row data + twiddle table into LDS (float4).
  const float4* xr4 = (const float4*)(x_re + row * NFFT);
  const float4* xi4 = (const float4*)(x_im + row * NFFT);
  const float4* wr4 = (const float4*)W_re;
  const float4* wi4 = (const float4*)W_im;
#pragma unroll
  for (int v = 0; v < 4; ++v) {
    const int i4 = tid + 256 * v;             // 1024 float4 = 4096 floats
    ((float4*)sAr)[i4] = xr4[i4];
    ((float4*)sAi)[i4] = xi4[i4];
    ((float4*)sWr)[i4] = wr4[i4];
    ((float4*)sWi)[i4] = wi4[i4];
  }
  __syncthreads();

  // Per-lane A-operand chunks of the exact 16-point DFT matrix:
  // F[M][K] = w16^(M*K) = W[(M*K*256) & 4095];  M = l15, K = 4c + 2*hi + v.
  v2f fa_r[4], fa_i[4], fa_n[4];
#pragma unroll
  for (int c = 0; c < 4; ++c) {
#pragma unroll
    for (int v = 0; v < 2; ++v) {
      const int k = 4 * c + 2 * hi + v;
      const int idx = (l15 * k * 256) & 4095;
      const float wr = sWr[idx], wi = sWi[idx];
      if (v == 0) { fa_r[c].x = wr; fa_i[c].x = wi; fa_n[c].x = -wi; }
      else        { fa_r[c].y = wr; fa_i[c].y = wi; fa_n[c].y = -wi; }
    }
  }

  // Three radix-16 passes, LDS ping-pong: A->B, B->A, A->B.
  fft_pass<1>(sAr, sAi, sBr, sBi, sWr, sWi, fa_r, fa_i, fa_n, wave, hi, l15);
  __syncthreads();
  fft_pass<2>(sBr, sBi, sAr, sAi, sWr, sWi, fa_r, fa_i, fa_n, wave, hi, l15);
  __syncthreads();
  fft_pass<3>(sAr, sAi, sBr, sBi, sWr, sWi, fa_r, fa_i, fa_n, wave, hi, l15);
  __syncthreads();

  // Result is in natural order in sB — coalesced float4 store.
  float4* yr4 = (float4*)(out + row * NFFT);
  float4* yi4 = (float4*)(out + im_off + row * NFFT);
#pragma unroll
  for (int v = 0; v < 4; ++v) {
    const int i4 = tid + 256 * v;
    yr4[i4] = ((float4*)sBr)[i4];
    yi4[i4] = ((float4*)sBi)[i4];
  }
}

extern "C" void kernel_launch(void* const* d_in, const int* in_sizes, int n_in,
                              void* d_out, int out_size, void* d_ws, size_t ws_size,
                              hipStream_t stream) {
  const float* x_re = (const float*)d_in[0];
  const float* x_im = (const float*)d_in[1];
  const float* W_re = (const float*)d_in[2];
  const float* W_im = (const float*)d_in[3];
  // d_in[4] = bitrev permutation — not needed (we compute the DFT directly).

  const int batch = in_sizes[0] / NFFT;                 // 8192
  const unsigned long long im_off = (unsigned long long)in_sizes[0]; // y_im offset in d_out

  const size_t shmem = 6 * NFFT * sizeof(float);        // 96 KB dynamic LDS
  hipLaunchKernelGGL(fft4096_wmma, dim3(batch), dim3(256), shmem, stream,
                     x_re, x_im, W_re, W_im, (float*)d_out, im_off);
}